// LM_MLE_with_iter_80410377716101
// MI455X (gfx1250) — compile-verified
//
#include <hip/hip_runtime.h>
#include <math.h>

// Problem constants (from reference)
#define RR     16
#define BTOT   16384
#define NPAR   4
#define NITER  12

#define INV2S2  0.2222222222222222f      // 1/(2*sigma^2), sigma=1.5
#define INV_S2  0.4444444444444444f      // 1/sigma^2
#define NORMC   0.07073553026306459f     // 1/(2*pi*sigma^2)
#define LAMBDA  0.01f
#define KEPS    2.220446049250313e-13f

typedef float v2f __attribute__((ext_vector_type(2)));
typedef float v8f __attribute__((ext_vector_type(8)));

// Fast reciprocal: single v_rcp_f32 (~1 ulp), avoids IEEE div expansion.
__device__ __forceinline__ float frcp(float v) { return __builtin_amdgcn_rcpf(v); }

// Separable PSF via rank-1 outer product on the matrix core:
//   psf(M,N) = [norm*exp(-(M+.5-x)^2/2s^2)] * [exp(-(N+.5-y)^2/2s^2)]
// A (16x4 f32): column K=0 = gx, rest zero.  B (4x16 f32): row K=0 = gy, rest zero.
// C (16x16 f32) layout: VGPR v, lane l -> M = v + 8*(l>>4), N = l & 15.
__device__ __forceinline__ v8f psf_outer(float x, float y, int lane) {
    float t  = (float)(lane & 15) + 0.5f;
    bool  lo = lane < 16;
    float c  = lo ? x : y;               // lanes 0-15 build gx, lanes 16-31 build gy
    float d  = t - c;
    float e  = __expf(-(d * d) * INV2S2);
    float gy = __shfl_xor(e, 16, 32);    // bring gy values down to lanes 0-15
    v2f a, b;
    a.x = lo ? (NORMC * e) : 0.0f;  a.y = 0.0f;   // K=1..3 columns of A are zero
    b.x = lo ? gy          : 0.0f;  b.y = 0.0f;   // K=1..3 rows of B are zero
    v8f cacc = {0.f,0.f,0.f,0.f,0.f,0.f,0.f,0.f};
    return __builtin_amdgcn_wmma_f32_16x16x4_f32(false, a, false, b,
                                                 (short)0, cacc, false, false);
}

__global__ __launch_bounds__(256) void lm_mle_kernel(
    const float* __restrict__ smp, const float* __restrict__ initial,
    const float* __restrict__ prm, const float* __restrict__ tol,
    float* __restrict__ out)
{
    const int lane = threadIdx.x & 31;
    const int b    = blockIdx.x * 8 + (threadIdx.x >> 5);   // one wave == one spot

    float* out_cur = out;                        // [B,4]
    float* out_ll  = out + BTOT * NPAR;          // [B]
    float* out_tr  = out + BTOT * NPAR + BTOT;   // [13,B,4]

    // ---- load samples once, clamped, in WMMA-C pixel order (stays in VGPRs) ----
    float s[8];
    const int base = b * 256 + (lane >> 4) * 128 + (lane & 15);
    #pragma unroll
    for (int v = 0; v < 8; ++v)
        s[v] = fmaxf(smp[base + v * 16], 1e-4f);

    // wave-uniform parameter state
    float x  = initial[b*4+0], y  = initial[b*4+1];
    float I  = initial[b*4+2], bg = initial[b*4+3];
    const float pmn[4] = { prm[0], prm[2], prm[4], prm[6] };
    const float pmx[4] = { prm[1], prm[3], prm[5], prm[7] };
    const float tl[4]  = { tol[0], tol[1], tol[2], tol[3] };

    // per-lane pixel coordinates under the C-layout
    const float Yn = (float)(lane & 15) + 0.5f;
    const float Xb = (float)((lane >> 4) * 8) + 0.5f;

    // traces bookkeeping: a == traces[(i-1)%13], b* == traces[i]
    float a0=0.f,a1=0.f,a2=0.f,a3=0.f;
    float b0=x, b1=y, b2=I, b3=bg;
    float mx=x, my=y, mI=I, mbg=bg;              // params of last good model eval
    int   good = 1;

    if (lane < 4)                                // traces[0] = cur0
        out_tr[b*4 + lane] = (lane==0) ? x : (lane==1) ? y : (lane==2) ? I : bg;

    for (int it = 0; it < NITER; ++it) {
        int goodu = __builtin_amdgcn_readfirstlane(good);  // scalar branch: EXEC stays all-1s
        if (goodu) {
            mx = x; my = y; mI = I; mbg = bg;              // freeze mu-params (pre-update cur)

            v8f psf8 = psf_outer(x, y, lane);

            float g0=0,g1=0,g2=0,g3=0;
            float h00=0,h01=0,h02=0,h03=0,h11=0,h12=0,h13=0,h22=0,h23=0,h33=0;
            const float dyn = Yn - y;
            #pragma unroll
            for (int v = 0; v < 8; ++v) {
                float psf = psf8[v];
                float mu  = fmaf(I, psf, bg);
                float mup = (mu > 0.f) ? mu : ((mu < 0.f) ? KEPS : 0.f);
                float r   = frcp(mup);                     // one v_rcp_f32
                float w   = (s[v] - mup) * r;
                float sr  = s[v] * r;
                float dw  = sr * r;
                float dxm = (Xb + (float)v) - x;
                float ip  = I * psf;
                float jx  = ip * dxm * INV_S2;
                float jy  = ip * dyn * INV_S2;
                g0 = fmaf(w, jx,  g0); g1 = fmaf(w, jy,  g1);
                g2 = fmaf(w, psf, g2); g3 += w;
                float t0 = dw * jx, t1 = dw * jy, t2 = dw * psf;
                h00 = fmaf(t0, jx,  h00); h01 = fmaf(t0, jy,  h01);
                h02 = fmaf(t0, psf, h02); h03 += t0;
                h11 = fmaf(t1, jy,  h11); h12 = fmaf(t1, psf, h12); h13 += t1;
                h22 = fmaf(t2, psf, h22); h23 += t2;
                h33 += dw;
            }

            // wave reduction (32 lanes) of 14 accumulators
            float acc[14] = {g0,g1,g2,g3,h00,h01,h02,h03,h11,h12,h13,h22,h23,h33};
            #pragma unroll
            for (int k = 0; k < 14; ++k) {
                float vv = acc[k];
                vv += __shfl_xor(vv, 1, 32);
                vv += __shfl_xor(vv, 2, 32);
                vv += __shfl_xor(vv, 4, 32);
                vv += __shfl_xor(vv, 8, 32);
                vv += __shfl_xor(vv, 16, 32);
                acc[k] = vv;
            }

            // Bmat = H + LAMBDA*diag(H)*I + 1e-6 I, with H = -sum(dw*ji*jj)
            float M[4][5];
            M[0][0] = -acc[4]*(1.f+LAMBDA) + 1e-6f; M[0][1] = -acc[5]; M[0][2] = -acc[6]; M[0][3] = -acc[7];
            M[1][0] = -acc[5]; M[1][1] = -acc[8]*(1.f+LAMBDA) + 1e-6f; M[1][2] = -acc[9]; M[1][3] = -acc[10];
            M[2][0] = -acc[6]; M[2][1] = -acc[9]; M[2][2] = -acc[11]*(1.f+LAMBDA) + 1e-6f; M[2][3] = -acc[12];
            M[3][0] = -acc[7]; M[3][1] = -acc[10]; M[3][2] = -acc[12]; M[3][3] = -acc[13]*(1.f+LAMBDA) + 1e-6f;
            M[0][4] = -acc[0]; M[1][4] = -acc[1]; M[2][4] = -acc[2]; M[3][4] = -acc[3];

            // Gauss-Jordan with stepwise conditional pivoting (fully unrolled, no dyn index)
            #pragma unroll
            for (int k = 0; k < 4; ++k) {
                #pragma unroll
                for (int r = 0; r < 4; ++r) {
                    if (r > k) {
                        bool sw = fabsf(M[r][k]) > fabsf(M[k][k]);
                        #pragma unroll
                        for (int j = 0; j < 5; ++j) {
                            float tk = M[k][j], tr = M[r][j];
                            M[k][j] = sw ? tr : tk;
                            M[r][j] = sw ? tk : tr;
                        }
                    }
                }
                float inv = frcp(M[k][k]);                 // one v_rcp_f32
                #pragma unroll
                for (int j = 0; j < 5; ++j) M[k][j] *= inv;
                #pragma unroll
                for (int r = 0; r < 4; ++r) {
                    if (r != k) {
                        float f = M[r][k];
                        #pragma unroll
                        for (int j = 0; j < 5; ++j) M[r][j] = fmaf(-f, M[k][j], M[r][j]);
                    }
                }
            }
            float cur[4] = { x, y, I, bg };
            float cn[4];
            #pragma unroll
            for (int p = 0; p < 4; ++p) {
                float dt = M[p][4];
                dt = isnan(dt) ? (-0.1f * cur[p]) : dt;
                cn[p] = fminf(fmaxf(cur[p] + dt, pmn[p]), pmx[p]);
            }

            if (lane < 4)                          // traces[it+1] = cur_new (good)
                out_tr[(it+1)*BTOT*NPAR + b*4 + lane] = cn[lane];

            // convergence: delta = |traces[it-1] - traces[it]|
            bool c0 = fabsf(a0 - b0) < tl[0], c1 = fabsf(a1 - b1) < tl[1];
            bool c2 = fabsf(a2 - b2) < tl[2], c3 = fabsf(a3 - b3) < tl[3];
            good = (c0 && c1 && c2 && c3) ? 0 : 1;

            a0 = b0; a1 = b1; a2 = b2; a3 = b3;
            b0 = cn[0]; b1 = cn[1]; b2 = cn[2]; b3 = cn[3];
            x = cn[0]; y = cn[1]; I = cn[2]; bg = cn[3];
        } else {
            if (lane < 4)                          // untouched traces rows stay zero
                out_tr[(it+1)*BTOT*NPAR + b*4 + lane] = 0.0f;
        }
    }

    // ---- log-likelihood from frozen mu parameters (second WMMA site) ----
    v8f psf8 = psf_outer(mx, my, lane);
    float ll = 0.0f;
    #pragma unroll
    for (int v = 0; v < 8; ++v) {
        float mu = fmaf(mI, psf8[v], mbg);
        ll += s[v] * __logf(mu * frcp(s[v])) - (mu - s[v]);
    }
    ll += __shfl_xor(ll, 1, 32);
    ll += __shfl_xor(ll, 2, 32);
    ll += __shfl_xor(ll, 4, 32);
    ll += __shfl_xor(ll, 8, 32);
    ll += __shfl_xor(ll, 16, 32);

    if (lane == 0) out_ll[b] = ll;
    if (lane < 4)
        out_cur[b*4 + lane] = (lane==0) ? x : (lane==1) ? y : (lane==2) ? I : bg;
}

extern "C" void kernel_launch(void* const* d_in, const int* in_sizes, int n_in,
                              void* d_out, int out_size, void* d_ws, size_t ws_size,
                              hipStream_t stream) {
    const float* smp     = (const float*)d_in[0];   // [B,16,16]
    const float* initial = (const float*)d_in[1];   // [B,4]
    const float* prm     = (const float*)d_in[2];   // [4,2]
    const float* tol     = (const float*)d_in[3];   // [4]
    float* out = (float*)d_out;                     // cur | loglik | traces
    (void)in_sizes; (void)n_in; (void)out_size; (void)d_ws; (void)ws_size;
    lm_mle_kernel<<<BTOT / 8, 256, 0, stream>>>(smp, initial, prm, tol, out);
}